// GCNAE_25907242729573
// MI455X (gfx1250) — compile-verified
//
#include <hip/hip_runtime.h>

#define N_NODES 100000
#define F_IN    512
#define F_HID   64
#define M_TILES (N_NODES / 16)   // 6250, exact

typedef float v2f __attribute__((ext_vector_type(2)));
typedef float v8f __attribute__((ext_vector_type(8)));

// ---------------- utility kernels ----------------

__global__ void GCN_zero_f32(float* __restrict__ p, long n) {
    long i = (long)blockIdx.x * blockDim.x + threadIdx.x;
    if (i < n) p[i] = 0.0f;
}

__global__ void GCN_degree(const int* __restrict__ src, const int* __restrict__ dst,
                           float* __restrict__ deg_out, float* __restrict__ deg_in, int E) {
    int e = blockIdx.x * blockDim.x + threadIdx.x;
    if (e < E) {
        atomicAdd(&deg_out[src[e]], 1.0f);
        atomicAdd(&deg_in[dst[e]], 1.0f);
    }
}

__global__ void GCN_norm(float* __restrict__ p, int n) {
    int i = blockIdx.x * blockDim.x + threadIdx.x;
    if (i < n) {
        float d = p[i];
        p[i] = (d > 0.0f) ? (1.0f / sqrtf(d)) : 1.0f;
    }
}

// edge scatter: out[dst[e]][c] += in[src[e]][c], 64 lanes per edge (coalesced 128B per half-wave)
__global__ void GCN_scatter_add64(const int* __restrict__ src, const int* __restrict__ dst,
                                  const float* __restrict__ in, float* __restrict__ out, int E) {
    long t = (long)blockIdx.x * blockDim.x + threadIdx.x;
    int e = (int)(t >> 6);
    int c = (int)(t & 63);
    if (e < E) {
        int s = src[e];
        int d = dst[e];
        atomicAdd(&out[(size_t)d * F_HID + c], in[(size_t)s * F_HID + c]);
    }
}

// h = relu(agg * norm_dst[row]) * norm_src[row]   (norm_src folded in for layer-2 gather)
__global__ void GCN_post_layer1(float* __restrict__ buf, const float* __restrict__ ndst,
                                const float* __restrict__ nsrc, long n) {
    long i = (long)blockIdx.x * blockDim.x + threadIdx.x;
    if (i < n) {
        long r = i >> 6;
        float v = buf[i] * ndst[r];
        v = v > 0.0f ? v : 0.0f;
        buf[i] = v * nsrc[r];
    }
}

// ---------------- WMMA GEMMs (f32 16x16x4) ----------------
// A operand layout: lane L holds A[M = L&15][K = k0 + 2*(L>>4) + {0,1}] in {a.x, a.y}
// B operand layout: lane L holds B[K = k0 + 2*(L>>4) + {0,1}][N = L&15] in {b.x, b.y}
// C/D layout:       lane L, vgpr v holds D[M = v + 8*(L>>4)][N = L&15]

// out[N,64] = (X[N,512] @ W1[512,64]) * norm_src[row]   -- one wave per 16x64 tile, 8 waves/block
__global__ void GCN_gemm1_wmma(const float* __restrict__ X, const float* __restrict__ W,
                               const float* __restrict__ nsrc, float* __restrict__ out,
                               int ntiles) {
    const int lane  = threadIdx.x & 31;
    const int wave  = threadIdx.x >> 5;
    const int tileM = blockIdx.x * 8 + wave;
    if (tileM >= ntiles) return;              // uniform per wave
    const int khalf = (lane >> 4) << 1;       // 0 or 2
    const int col   = lane & 15;
    const int row   = tileM * 16 + (lane & 15);
    const float* __restrict__ xrow = X + (size_t)row * F_IN;

    v8f acc0 = {}, acc1 = {}, acc2 = {}, acc3 = {};
    for (int k = 0; k < F_IN; k += 4) {
        v2f a;
        a.x = xrow[k + khalf];
        a.y = xrow[k + khalf + 1];
        const float* __restrict__ w0 = W + (size_t)(k + khalf) * F_HID + col;
        const float* __restrict__ w1 = w0 + F_HID;
        v2f b0, b1, b2, b3;
        b0.x = w0[0];  b0.y = w1[0];
        b1.x = w0[16]; b1.y = w1[16];
        b2.x = w0[32]; b2.y = w1[32];
        b3.x = w0[48]; b3.y = w1[48];
        acc0 = __builtin_amdgcn_wmma_f32_16x16x4_f32(false, a, false, b0, (short)0, acc0, false, false);
        acc1 = __builtin_amdgcn_wmma_f32_16x16x4_f32(false, a, false, b1, (short)0, acc1, false, false);
        acc2 = __builtin_amdgcn_wmma_f32_16x16x4_f32(false, a, false, b2, (short)0, acc2, false, false);
        acc3 = __builtin_amdgcn_wmma_f32_16x16x4_f32(false, a, false, b3, (short)0, acc3, false, false);
    }
    const int rbase = tileM * 16 + ((lane >> 4) << 3);
    const int n     = lane & 15;
    for (int v = 0; v < 8; ++v) {
        int r = rbase + v;
        float s = nsrc[r];
        float* __restrict__ orow = out + (size_t)r * F_HID + n;
        orow[0]  = acc0[v] * s;
        orow[16] = acc1[v] * s;
        orow[32] = acc2[v] * s;
        orow[48] = acc3[v] * s;
    }
}

// out[N,512] = (H[N,64] * norm_dst[row]) @ W2[64,512]  -- 8 waves/block, each wave a 16x64 slice
__global__ void GCN_gemm2_wmma(const float* __restrict__ H, const float* __restrict__ W,
                               const float* __restrict__ ndst, float* __restrict__ out) {
    const int lane  = threadIdx.x & 31;
    const int wave  = threadIdx.x >> 5;   // column block 0..7
    const int tileM = blockIdx.x;
    const int cbase = wave * 64;
    const int khalf = (lane >> 4) << 1;
    const int col   = lane & 15;
    const int row   = tileM * 16 + (lane & 15);
    const float sA  = ndst[row];
    const float* __restrict__ hrow = H + (size_t)row * F_HID;

    v8f acc0 = {}, acc1 = {}, acc2 = {}, acc3 = {};
    for (int k = 0; k < F_HID; k += 4) {
        v2f a;
        a.x = hrow[k + khalf] * sA;
        a.y = hrow[k + khalf + 1] * sA;
        const float* __restrict__ w0 = W + (size_t)(k + khalf) * F_IN + cbase + col;
        const float* __restrict__ w1 = w0 + F_IN;
        v2f b0, b1, b2, b3;
        b0.x = w0[0];  b0.y = w1[0];
        b1.x = w0[16]; b1.y = w1[16];
        b2.x = w0[32]; b2.y = w1[32];
        b3.x = w0[48]; b3.y = w1[48];
        acc0 = __builtin_amdgcn_wmma_f32_16x16x4_f32(false, a, false, b0, (short)0, acc0, false, false);
        acc1 = __builtin_amdgcn_wmma_f32_16x16x4_f32(false, a, false, b1, (short)0, acc1, false, false);
        acc2 = __builtin_amdgcn_wmma_f32_16x16x4_f32(false, a, false, b2, (short)0, acc2, false, false);
        acc3 = __builtin_amdgcn_wmma_f32_16x16x4_f32(false, a, false, b3, (short)0, acc3, false, false);
    }
    const int rbase = tileM * 16 + ((lane >> 4) << 3);
    const int n     = lane & 15;
    for (int v = 0; v < 8; ++v) {
        float* __restrict__ orow = out + (size_t)(rbase + v) * F_IN + cbase + n;
        orow[0]  = acc0[v];
        orow[16] = acc1[v];
        orow[32] = acc2[v];
        orow[48] = acc3[v];
    }
}

// ---------------- launch ----------------

extern "C" void kernel_launch(void* const* d_in, const int* in_sizes, int n_in,
                              void* d_out, int out_size, void* d_ws, size_t ws_size,
                              hipStream_t stream) {
    const float* X   = (const float*)d_in[0];
    const int*   src = (const int*)d_in[1];
    const int*   dst = (const int*)d_in[2];
    const float* W1  = (const float*)d_in[3];
    const float* W2  = (const float*)d_in[4];
    float*       out = (float*)d_out;
    const int E = in_sizes[1];

    float* nsrc = (float*)d_ws;
    float* ndst = nsrc + N_NODES;
    float* bufA = ndst + N_NODES;                      // N x 64
    float* bufB = bufA + (size_t)N_NODES * F_HID;      // N x 64

    const long nhid = (long)N_NODES * F_HID;

    // degrees -> norms
    GCN_zero_f32<<<(2 * N_NODES + 255) / 256, 256, 0, stream>>>(nsrc, 2L * N_NODES);
    GCN_degree<<<(E + 255) / 256, 256, 0, stream>>>(src, dst, nsrc, ndst, E);
    GCN_norm<<<(N_NODES + 255) / 256, 256, 0, stream>>>(nsrc, N_NODES);
    GCN_norm<<<(N_NODES + 255) / 256, 256, 0, stream>>>(ndst, N_NODES);

    // layer 1: bufA = (X @ W1) * norm_src
    GCN_gemm1_wmma<<<(M_TILES + 7) / 8, 256, 0, stream>>>(X, W1, nsrc, bufA, M_TILES);

    // layer 1 aggregation: bufB = segment_sum(bufA[src], dst)
    GCN_zero_f32<<<(unsigned)((nhid + 255) / 256), 256, 0, stream>>>(bufB, nhid);
    GCN_scatter_add64<<<(unsigned)(((long)E * 64 + 255) / 256), 256, 0, stream>>>(src, dst, bufA, bufB, E);

    // h = relu(bufB * norm_dst) * norm_src  (pre-scaled for layer-2 gather)
    GCN_post_layer1<<<(unsigned)((nhid + 255) / 256), 256, 0, stream>>>(bufB, ndst, nsrc, nhid);

    // layer 2 aggregation: bufA = segment_sum(bufB[src], dst)
    GCN_zero_f32<<<(unsigned)((nhid + 255) / 256), 256, 0, stream>>>(bufA, nhid);
    GCN_scatter_add64<<<(unsigned)(((long)E * 64 + 255) / 256), 256, 0, stream>>>(src, dst, bufB, bufA, E);

    // out = (bufA * norm_dst) @ W2
    GCN_gemm2_wmma<<<M_TILES, 256, 0, stream>>>(bufA, W2, ndst, out);
}